// saltpepper_new_57543971831865
// MI455X (gfx1250) — compile-verified
//
#include <hip/hip_runtime.h>

// out[c,p] = relu(x[c,p] + mask[p]) + plus_one[p]   (elementwise, broadcast over C)
// Pure streaming kernel: HBM-bound (537 MB traffic, ~23 us floor @ 23.3 TB/s).

typedef __attribute__((ext_vector_type(4))) float v4f;

__device__ __forceinline__ v4f relu_add(v4f xv, v4f m, v4f po) {
    v4f r;
    r.x = fmaxf(xv.x + m.x, 0.0f) + po.x;
    r.y = fmaxf(xv.y + m.y, 0.0f) + po.y;
    r.z = fmaxf(xv.z + m.z, 0.0f) + po.z;
    r.w = fmaxf(xv.w + m.w, 0.0f) + po.w;
    return r;
}

// Specialized C==3 path: one thread owns 4 pixels x 3 channels.
// mask/plus_one read ONCE from HBM, reused from registers for all channels.
__global__ __launch_bounds__(256) void saltpepper_c3_kernel(
    const float* __restrict__ x,
    const float* __restrict__ mask,
    const float* __restrict__ plus_one,
    float* __restrict__ out,
    int n4, int hw)
{
    int p4 = blockIdx.x * blockDim.x + threadIdx.x;
    if (p4 >= n4) return;
    size_t p  = (size_t)p4 * 4;
    size_t HW = (size_t)hw;

    // Issue all 5 x b128 loads up front (5 outstanding LOADcnt) — NT policy:
    // 537 MB stream > 192 MB L2, no reuse beyond registers.
    const v4f m  = __builtin_nontemporal_load((const v4f*)(mask     + p));
    const v4f po = __builtin_nontemporal_load((const v4f*)(plus_one + p));
    const v4f x0 = __builtin_nontemporal_load((const v4f*)(x            + p));
    const v4f x1 = __builtin_nontemporal_load((const v4f*)(x + HW       + p));
    const v4f x2 = __builtin_nontemporal_load((const v4f*)(x + 2 * HW   + p));

    v4f r0 = relu_add(x0, m, po);
    v4f r1 = relu_add(x1, m, po);
    v4f r2 = relu_add(x2, m, po);

    __builtin_nontemporal_store(r0, (v4f*)(out          + p));
    __builtin_nontemporal_store(r1, (v4f*)(out + HW     + p));
    __builtin_nontemporal_store(r2, (v4f*)(out + 2 * HW + p));
}

// Generic fallback: arbitrary channel count, vectorized over pixels.
__global__ __launch_bounds__(256) void saltpepper_generic_kernel(
    const float* __restrict__ x,
    const float* __restrict__ mask,
    const float* __restrict__ plus_one,
    float* __restrict__ out,
    int n4, int hw, int nchan)
{
    int p4 = blockIdx.x * blockDim.x + threadIdx.x;
    if (p4 >= n4) return;
    size_t p  = (size_t)p4 * 4;
    size_t HW = (size_t)hw;

    const v4f m  = __builtin_nontemporal_load((const v4f*)(mask     + p));
    const v4f po = __builtin_nontemporal_load((const v4f*)(plus_one + p));
    for (int c = 0; c < nchan; ++c) {
        const v4f xv = __builtin_nontemporal_load((const v4f*)(x + (size_t)c * HW + p));
        __builtin_nontemporal_store(relu_add(xv, m, po),
                                    (v4f*)(out + (size_t)c * HW + p));
    }
}

// Scalar tail kernel for hw not divisible by 4 (not hit for 4096x4096, but safe).
__global__ __launch_bounds__(256) void saltpepper_tail_kernel(
    const float* __restrict__ x,
    const float* __restrict__ mask,
    const float* __restrict__ plus_one,
    float* __restrict__ out,
    int tail_start, int hw, int nchan)
{
    int p = tail_start + blockIdx.x * blockDim.x + threadIdx.x;
    if (p >= hw) return;
    float m  = mask[p];
    float po = plus_one[p];
    for (int c = 0; c < nchan; ++c) {
        size_t idx = (size_t)c * hw + p;
        out[idx] = fmaxf(x[idx] + m, 0.0f) + po;
    }
}

extern "C" void kernel_launch(void* const* d_in, const int* in_sizes, int n_in,
                              void* d_out, int out_size, void* d_ws, size_t ws_size,
                              hipStream_t stream) {
    const float* x        = (const float*)d_in[0];
    const float* mask     = (const float*)d_in[1];
    const float* plus_one = (const float*)d_in[2];
    float* out            = (float*)d_out;

    const int hw    = in_sizes[1];            // 1*1*H*W broadcast plane
    const int nchan = in_sizes[0] / hw;       // 3
    const int n4    = hw / 4;                 // float4 count per plane
    const int tail  = n4 * 4;                 // first non-vectorized pixel

    const int block = 256;
    if (n4 > 0) {
        const int grid = (n4 + block - 1) / block;
        if (nchan == 3) {
            saltpepper_c3_kernel<<<grid, block, 0, stream>>>(
                x, mask, plus_one, out, n4, hw);
        } else {
            saltpepper_generic_kernel<<<grid, block, 0, stream>>>(
                x, mask, plus_one, out, n4, hw, nchan);
        }
    }
    if (tail < hw) {
        const int ntail = hw - tail;
        const int grid  = (ntail + block - 1) / block;
        saltpepper_tail_kernel<<<grid, block, 0, stream>>>(
            x, mask, plus_one, out, tail, hw, nchan);
    }
}